// PolicyGradientLoss_36747740185105
// MI455X (gfx1250) — compile-verified
//
#include <hip/hip_runtime.h>
#include <hip/hip_bf16.h>

// Problem constants (from reference)
#define BQ 512
#define SQ 512
#define AQ 200
#define LOSS_CUTOFF 100000.0f
#define EPS_F 1e-8f

typedef __attribute__((ext_vector_type(2))) float v2f;
typedef __attribute__((ext_vector_type(8))) float v8f;

// ---------------------------------------------------------------------------
// Kernel 1: one block per batch row b. 256 threads = 8 wave32s.
// Streams logits[b, :, :] once (coalesced float4 loads), computes per-(b,s)
// logsumexp + gathered action logit fully in-wave via shfl butterflies,
// accumulates total_logp[b] and total_rewards[b] into workspace.
// ---------------------------------------------------------------------------
__global__ __launch_bounds__(256) void pg_main_kernel(
    const float* __restrict__ logits,
    const int*   __restrict__ actions,
    const float* __restrict__ rewards,
    const float* __restrict__ terminals,
    float* __restrict__ tlogp,   // [BQ]
    float* __restrict__ trew)    // [BQ]
{
    const int b    = blockIdx.x;
    const int tid  = threadIdx.x;
    const int lane = tid & 31;
    const int wave = tid >> 5;          // 0..7

    __shared__ float red[16];

    const long long rowbase = (long long)b * SQ;

    float acc_logp = 0.0f;

    const int a0   = lane * 4;          // 0..124
    const int a1   = a0 + 128;          // 128..252
    const bool has1 = (a1 < AQ);        // lanes 0..17

    for (int s = wave; s < SQ; s += 8) {
        const float* row = logits + (rowbase + s) * (long long)AQ;

        // Prefetch this wave's next row (s+8) while we reduce the current one.
        __builtin_prefetch(row + 8 * AQ + lane * 16, 0, 0);

        float4 v0 = *(const float4*)(row + a0);
        float4 v1 = make_float4(0.f, 0.f, 0.f, 0.f);
        float vmax = fmaxf(fmaxf(v0.x, v0.y), fmaxf(v0.z, v0.w));
        if (has1) {
            v1 = *(const float4*)(row + a1);
            vmax = fmaxf(vmax, fmaxf(fmaxf(v1.x, v1.y), fmaxf(v1.z, v1.w)));
        }
        // wave32 butterfly max -> uniform across lanes
        #pragma unroll
        for (int off = 16; off > 0; off >>= 1)
            vmax = fmaxf(vmax, __shfl_xor(vmax, off, 32));

        float se = __expf(v0.x - vmax) + __expf(v0.y - vmax) +
                   __expf(v0.z - vmax) + __expf(v0.w - vmax);
        if (has1)
            se += __expf(v1.x - vmax) + __expf(v1.y - vmax) +
                  __expf(v1.z - vmax) + __expf(v1.w - vmax);
        #pragma unroll
        for (int off = 16; off > 0; off >>= 1)
            se += __shfl_xor(se, off, 32);

        // Gather logits[b,s,action] from whichever lane holds it.
        const int act = actions[rowbase + s];
        float g = 0.0f;
        if (act >= a0 && act < a0 + 4) g = ((const float*)&v0)[act - a0];
        if (has1 && act >= a1 && act < a1 + 4) g = ((const float*)&v1)[act - a1];
        #pragma unroll
        for (int off = 16; off > 0; off >>= 1)
            g += __shfl_xor(g, off, 32);

        const float lse = vmax + __logf(se);
        const float t   = terminals[rowbase + s];
        // total_logp accumulates -(g - lse)*(1-t) = (lse - g)*(1-t)
        acc_logp += (lse - g) * (1.0f - t);
    }

    if (lane == 0) red[wave] = acc_logp;   // wave-uniform after butterflies

    // total_rewards[b]: each thread sums 2 of the 512 entries, reduce.
    float r = rewards[rowbase + tid] + rewards[rowbase + tid + 256];
    #pragma unroll
    for (int off = 16; off > 0; off >>= 1)
        r += __shfl_xor(r, off, 32);
    if (lane == 0) red[8 + wave] = r;

    __syncthreads();
    if (tid == 0) {
        float tl = 0.0f, tr = 0.0f;
        #pragma unroll
        for (int i = 0; i < 8; ++i) { tl += red[i]; tr += red[8 + i]; }
        tlogp[b] = tl;
        trew[b]  = tr;
    }
}

// ---------------------------------------------------------------------------
// Kernel 2: single wave32 (EXEC all ones, as WMMA requires). The three
// length-512 reductions are done as matrix x ones with V_WMMA_F32_16X16X4_F32:
// A = 16x4 data tile, B = all-ones, C accumulates 8 tiles. D[m][n] is then the
// accumulated row-sum (independent of n); fold the 16 rows per the documented
// C/D layout (VGPR r <-> M=r on lanes 0-15, M=8+r on lanes 16-31).
// ---------------------------------------------------------------------------
__global__ __launch_bounds__(32) void pg_final_kernel(
    const float* __restrict__ tlogp,
    const float* __restrict__ trew,
    float* __restrict__ out)
{
    const int lane = threadIdx.x;       // 0..31, exactly one wave
    const int m = lane & 15;
    const int h = lane >> 4;            // 0: K={0,1}, 1: K={2,3}

    v8f cp = {};   // masked prod sums
    v8f cc = {};   // mask count
    v8f ca = {};   // |total_rewards| sums

    v2f ones; ones[0] = 1.0f; ones[1] = 1.0f;   // B = all-ones (layout-proof)

    #pragma unroll
    for (int chunk = 0; chunk < 8; ++chunk) {
        // A-matrix 16x4 f32 layout: lane = M + 16*(K>=2), VGPR = K&1.
        const int e = chunk * 64 + m * 4 + h * 2;
        const float tl0 = tlogp[e],  tl1 = tlogp[e + 1];
        const float tr0 = trew[e],   tr1 = trew[e + 1];
        const bool  m0 = tl0 < LOSS_CUTOFF;
        const bool  m1 = tl1 < LOSS_CUTOFF;

        v2f ap, ac, aa;
        ap[0] = m0 ? tl0 * tr0 : 0.0f;  ap[1] = m1 ? tl1 * tr1 : 0.0f;
        ac[0] = m0 ? 1.0f : 0.0f;       ac[1] = m1 ? 1.0f : 0.0f;
        aa[0] = fabsf(tr0);             aa[1] = fabsf(tr1);

        cp = __builtin_amdgcn_wmma_f32_16x16x4_f32(false, ap, false, ones,
                                                   (short)0, cp, false, false);
        cc = __builtin_amdgcn_wmma_f32_16x16x4_f32(false, ac, false, ones,
                                                   (short)0, cc, false, false);
        ca = __builtin_amdgcn_wmma_f32_16x16x4_f32(false, aa, false, ones,
                                                   (short)0, ca, false, false);
    }

    // Fold D rows: lanes 0-15 hold M=0..7 in VGPRs 0..7, lanes 16-31 hold 8..15.
    float sp = 0.0f, sc = 0.0f, sa = 0.0f;
    #pragma unroll
    for (int i = 0; i < 8; ++i) { sp += cp[i]; sc += cc[i]; sa += ca[i]; }
    sp += __shfl_xor(sp, 16, 32);
    sc += __shfl_xor(sc, 16, 32);
    sa += __shfl_xor(sa, 16, 32);

    if (lane == 0) {
        const float rewardloss_mean = sp / sc;
        out[0] = rewardloss_mean / (sa * (1.0f / (float)BQ) + EPS_F);
    }
}

// ---------------------------------------------------------------------------
extern "C" void kernel_launch(void* const* d_in, const int* in_sizes, int n_in,
                              void* d_out, int out_size, void* d_ws, size_t ws_size,
                              hipStream_t stream) {
    (void)in_sizes; (void)n_in; (void)out_size; (void)ws_size;

    const float* logits    = (const float*)d_in[0];
    const int*   actions   = (const int*)  d_in[1];
    const float* rewards   = (const float*)d_in[2];
    const float* terminals = (const float*)d_in[3];

    float* ws    = (float*)d_ws;
    float* tlogp = ws;          // [BQ]
    float* trew  = ws + BQ;     // [BQ]

    pg_main_kernel<<<BQ, 256, 0, stream>>>(logits, actions, rewards, terminals,
                                           tlogp, trew);
    pg_final_kernel<<<1, 32, 0, stream>>>(tlogp, trew, (float*)d_out);
}